// NavierStokesFieldEvolution_30571577213323
// MI455X (gfx1250) — compile-verified
//
#include <hip/hip_runtime.h>

typedef float v2f __attribute__((ext_vector_type(2)));
typedef float v8f __attribute__((ext_vector_type(8)));

#define NSIDE 256
#define NBATCH 8
#define PLANE_ELEMS (NBATCH * NSIDE * NSIDE)   // 524288 floats = 2 MB per plane

__device__ __forceinline__ v8f wmma4(v2f a, v2f b, v8f c) {
  // D = A(16x4) * B(4x16) + C(16x16), f32. 4 chained calls give K=16.
  return __builtin_amdgcn_wmma_f32_16x16x4_f32(false, a, false, b, (short)0, c, false, false);
}

struct FieldPtrs { float* re[4]; float* im[4]; };

// Each wave processes `lpw` 256-point complex DFT lines (four-step radix-16 via WMMA).
// astride: element stride along transform axis, ostride: stride between lines.
// In-place. sign=-1 forward, +1 inverse; scale applied at final store.
// Grid must satisfy: gridDim.x * 8 * lpw == nfields * NBATCH * 256.
__global__ __launch_bounds__(256) void dft256_pass(FieldPtrs fp, int astride, int ostride,
                                                   float sign, float scale, int lpw) {
  __shared__ float lre[8][256];
  __shared__ float lim[8][256];
  const int wave = threadIdx.x >> 5;
  const int lane = threadIdx.x & 31;
  const int wgid = blockIdx.x * 8 + wave;          // global wave id
  const int nwaves = gridDim.x * 8;

  const int n  = lane & 15;        // A-layout M index / B-layout N index
  const int hi = lane >> 4;
  const int khalf = hi << 1;       // K offset 0 or 2 within a 4-chunk

  // Constant DFT-16 operand in A layout: F[m][k] = exp(sign*2*pi*i*m*k/16)
  v2f Ar[4], Ai[4], Ani[4];
#pragma unroll
  for (int c = 0; c < 4; ++c) {
#pragma unroll
    for (int j = 0; j < 2; ++j) {
      int k = 4 * c + j + khalf;
      float ang = sign * 0.39269908169872414f * (float)((n * k) & 15);
      float s, co; __sincosf(ang, &s, &co);
      Ar[c][j] = co; Ai[c][j] = s; Ani[c][j] = -s;
    }
  }
  // Twiddle for C-layout element (M=k1, N=n0): exp(sign*2*pi*i*M*N/256)
  float twr[8], twi[8];
#pragma unroll
  for (int v = 0; v < 8; ++v) {
    int Mv = v + 8 * hi;
    float ang = sign * 0.024543692606170259f * (float)((Mv * n) & 255);
    __sincosf(ang, &twi[v], &twr[v]);
  }

  for (int it = 0; it < lpw; ++it) {
    const int gl  = wgid + it * nwaves;             // global line id
    const int f   = gl >> 11;                       // / (NBATCH*256)
    const int rem = gl & 2047;
    const int b   = rem >> 8;
    const int o   = rem & 255;
    float* pre = fp.re[f] + (size_t)b * (NSIDE * NSIDE);
    float* pim = fp.im[f] + (size_t)b * (NSIDE * NSIDE);

    // Stage 1: B element (K=n1, N=n0) = x[16*n1 + n0]
    v2f Br[4], Bi[4];
#pragma unroll
    for (int c = 0; c < 4; ++c) {
#pragma unroll
      for (int j = 0; j < 2; ++j) {
        int k = 4 * c + j + khalf;
        int idx = o * ostride + (16 * k + n) * astride;
        Br[c][j] = pre[idx];
        Bi[c][j] = pim[idx];
      }
    }
    v8f Cr = {}; v8f Ci = {};
#pragma unroll
    for (int c = 0; c < 4; ++c) Cr = wmma4(Ar[c],  Br[c], Cr);
#pragma unroll
    for (int c = 0; c < 4; ++c) Cr = wmma4(Ani[c], Bi[c], Cr);
#pragma unroll
    for (int c = 0; c < 4; ++c) Ci = wmma4(Ai[c],  Br[c], Ci);
#pragma unroll
    for (int c = 0; c < 4; ++c) Ci = wmma4(Ar[c],  Bi[c], Ci);

    // Twiddle, then transpose through wave-private LDS tile.
#pragma unroll
    for (int v = 0; v < 8; ++v) {
      int Mv = v + 8 * hi;
      float cr = Cr[v], ci = Ci[v];
      lre[wave][Mv * 16 + n] = cr * twr[v] - ci * twi[v];
      lim[wave][Mv * 16 + n] = cr * twi[v] + ci * twr[v];
    }
    __syncthreads();

    // Stage 2: E = F * C'^T ; B element (K=m, N=p) = tile[p][m]
#pragma unroll
    for (int c = 0; c < 4; ++c) {
#pragma unroll
      for (int j = 0; j < 2; ++j) {
        int k = 4 * c + j + khalf;
        Br[c][j] = lre[wave][n * 16 + k];
        Bi[c][j] = lim[wave][n * 16 + k];
      }
    }
    v8f Dr = {}; v8f Di = {};
#pragma unroll
    for (int c = 0; c < 4; ++c) Dr = wmma4(Ar[c],  Br[c], Dr);
#pragma unroll
    for (int c = 0; c < 4; ++c) Dr = wmma4(Ani[c], Bi[c], Dr);
#pragma unroll
    for (int c = 0; c < 4; ++c) Di = wmma4(Ai[c],  Br[c], Di);
#pragma unroll
    for (int c = 0; c < 4; ++c) Di = wmma4(Ar[c],  Bi[c], Di);

    // Store: E(M=k0, N=k1) -> X[k1 + 16*k0] = index N + 16*M
#pragma unroll
    for (int v = 0; v < 8; ++v) {
      int Mv = v + 8 * hi;
      int idx = o * ostride + (n + 16 * Mv) * astride;
      pre[idx] = scale * Dr[v];
      pim[idx] = scale * Di[v];
    }
    __syncthreads();   // protect LDS tile before next iteration overwrites it
  }
}

__global__ __launch_bounds__(256) void load_real_kernel(const float* __restrict__ src,
                                                        float* __restrict__ re, float* __restrict__ im) {
  int t = blockIdx.x * 256 + threadIdx.x;
  if (t >= PLANE_ELEMS) return;
  re[t] = src[t];
  im[t] = 0.0f;
}

__global__ __launch_bounds__(256) void copy_real_kernel(const float* __restrict__ re,
                                                        float* __restrict__ out) {
  int t = blockIdx.x * 256 + threadIdx.x;
  if (t >= PLANE_ELEMS) return;
  out[t] = re[t];
}

// psi = -u/lap_nz ; vxh = i*ky*psi ; vyh = -i*kx*psi ; gxh = i*kx*u ; gyh = i*ky*u
__global__ __launch_bounds__(256) void prep_kernel(const float* __restrict__ u_re, const float* __restrict__ u_im,
    float* vxr, float* vxi, float* vyr, float* vyi,
    float* gxr, float* gxi, float* gyr, float* gyi) {
  int t = blockIdx.x * 256 + threadIdx.x;
  if (t >= PLANE_ELEMS) return;
  int r = t & 65535;
  int i = r >> 8, j = r & 255;
  float kx = (float)((i < 128) ? i : i - 256);
  float ky = (float)((j < 128) ? j : j - 256);
  float k2 = kx * kx + ky * ky;
  float lapnz = (r == 0) ? 1.0f : -k2;
  float ur = u_re[t], ui = u_im[t];
  float pr = -ur / lapnz, pi = -ui / lapnz;
  vxr[t] = -ky * pi; vxi[t] =  ky * pr;
  vyr[t] =  kx * pi; vyi[t] = -kx * pr;
  gxr[t] = -kx * ui; gxi[t] =  kx * ur;
  gyr[t] = -ky * ui; gyi[t] =  ky * ur;
}

// n = -(vx*gx + vy*gy), written in-place into vx planes (im cleared for forward FFT)
__global__ __launch_bounds__(256) void nonlinear_kernel(float* __restrict__ vxr, const float* __restrict__ vyr,
                                                        const float* __restrict__ gxr, const float* __restrict__ gyr,
                                                        float* __restrict__ n_im) {
  int t = blockIdx.x * 256 + threadIdx.x;
  if (t >= PLANE_ELEMS) return;
  float v = -(vxr[t] * gxr[t] + vyr[t] * gyr[t]);
  vxr[t] = v;
  n_im[t] = 0.0f;
}

// h = adv*filt + f_hat + beta*h ; u = (u + gdt*h + mu*lin*u)/(1 - mu*lin)
__global__ __launch_bounds__(256) void update_kernel(const float* __restrict__ adv_re, const float* __restrict__ adv_im,
                                                     float* u_re, float* u_im, float* h_re, float* h_im,
                                                     float beta, float gdt, float mu) {
  int t = blockIdx.x * 256 + threadIdx.x;
  if (t >= PLANE_ELEMS) return;
  int r = t & 65535;
  int i = r >> 8, j = r & 255;
  // 2/3-rule filter (exact translation of reference rfft-space mask + hermitian image)
  bool keep;
  if (j <= 85)        keep = (i <= 84) || (i >= 171);
  else if (j >= 171)  keep = (i <= 85) || (i >= 172);
  else                keep = false;
  float ar = keep ? adv_re[t] : 0.0f;
  float ai = keep ? adv_im[t] : 0.0f;
  // f_hat: nonzero only at (kx=0, ky=+-4): -2*N^2*exp(+-i*pi/64)
  if (i == 0 && (j == 4 || j == 252)) {
    const float FHR = -131072.0f * 0.9987954562f;  // -2N^2 cos(pi/64)
    const float FHI = -131072.0f * 0.0490676743f;  // -2N^2 sin(pi/64)
    ar += FHR;
    ai += (j == 4) ? FHI : -FHI;
  }
  float hr = ar + beta * h_re[t];
  float hi = ai + beta * h_im[t];
  float kx = (float)((i < 128) ? i : i - 256);
  float ky = (float)((j < 128) ? j : j - 256);
  float k2 = kx * kx + ky * ky;
  float lin = -0.001f * k2 - 0.1f;          // VISC*lap - DRAG
  float ml = mu * lin;
  float den = 1.0f - ml;
  float ur = u_re[t], ui = u_im[t];
  u_re[t] = (ur + gdt * hr + ml * ur) / den;
  u_im[t] = (ui + gdt * hi + ml * ui) / den;
  h_re[t] = hr;
  h_im[t] = hi;
}

extern "C" void kernel_launch(void* const* d_in, const int* in_sizes, int n_in,
                              void* d_out, int out_size, void* d_ws, size_t ws_size,
                              hipStream_t stream) {
  (void)in_sizes; (void)n_in; (void)out_size; (void)ws_size;
  const float* vort = (const float*)d_in[0];
  float* out = (float*)d_out;
  float* w = (float*)d_ws;
  const size_t P = (size_t)PLANE_ELEMS;
  // Workspace: 12 planes * 2MB = 24MB
  float* u_re = w + 0 * P; float* u_im = w + 1 * P;
  float* h_re = w + 2 * P; float* h_im = w + 3 * P;
  float* f_re[4]; float* f_im[4];
  for (int q = 0; q < 4; ++q) { f_re[q] = w + (4 + 2 * q) * P; f_im[q] = w + (5 + 2 * q) * P; }

  const dim3 pw(256), pg((PLANE_ELEMS + 255) / 256);
  const float INV = 1.0f / 256.0f;   // per-pass inverse scale -> 1/N^2 total

  // Pass grids: lines = nfields*NBATCH*256 ; blocks = lines / (8 waves * lpw)
  const int LPW1 = 2, G1 = (1 * NBATCH * 256) / (8 * LPW1);   // 128 blocks, single field
  const int LPW4 = 4, G4 = (4 * NBATCH * 256) / (8 * LPW4);   // 256 blocks, 4 fields

  FieldPtrs fu{}; fu.re[0] = u_re; fu.im[0] = u_im;
  FieldPtrs fa{}; for (int q = 0; q < 4; ++q) { fa.re[q] = f_re[q]; fa.im[q] = f_im[q]; }
  FieldPtrs fn{}; fn.re[0] = f_re[0]; fn.im[0] = f_im[0];
  FieldPtrs fs{}; fs.re[0] = f_re[1]; fs.im[0] = f_im[1];

  // w0 = fft2(vort_init)
  load_real_kernel<<<pg, pw, 0, stream>>>(vort, u_re, u_im);
  dft256_pass<<<G1, 256, 0, stream>>>(fu, 1, 256, -1.0f, 1.0f, LPW1);   // along axis1
  dft256_pass<<<G1, 256, 0, stream>>>(fu, 256, 1, -1.0f, 1.0f, LPW1);   // along axis0

  const double AL[6] = {0.0, 0.1496590219993, 0.3704009573644, 0.6222557631345, 0.9582821306748, 1.0};
  const double BE[5] = {0.0, -0.4178904745, -1.192151694643, -1.697784692471, -1.514183444257};
  const double GA[5] = {0.1496590219993, 0.3792103129999, 0.8229550293869, 0.6994504559488, 0.1530572479681};
  const double DT = 0.001;

  for (int step = 0; step < 90; ++step) {
    for (int k = 0; k < 5; ++k) {
      prep_kernel<<<pg, pw, 0, stream>>>(u_re, u_im,
          f_re[0], f_im[0], f_re[1], f_im[1], f_re[2], f_im[2], f_re[3], f_im[3]);
      // inverse FFT2 of the 4 spectral fields (batched: 4*8*256 lines)
      dft256_pass<<<G4, 256, 0, stream>>>(fa, 1, 256, 1.0f, INV, LPW4);
      dft256_pass<<<G4, 256, 0, stream>>>(fa, 256, 1, 1.0f, INV, LPW4);
      nonlinear_kernel<<<pg, pw, 0, stream>>>(f_re[0], f_re[1], f_re[2], f_re[3], f_im[0]);
      // forward FFT2 of n
      dft256_pass<<<G1, 256, 0, stream>>>(fn, 1, 256, -1.0f, 1.0f, LPW1);
      dft256_pass<<<G1, 256, 0, stream>>>(fn, 256, 1, -1.0f, 1.0f, LPW1);
      float beta = (float)BE[k];
      float gdt  = (float)(GA[k] * DT);
      float mu   = (float)(0.5 * DT * (AL[k + 1] - AL[k]));
      update_kernel<<<pg, pw, 0, stream>>>(f_re[0], f_im[0], u_re, u_im, h_re, h_im, beta, gdt, mu);
    }
    if (step % 30 == 29) {
      int rec = step / 30;
      hipMemcpyAsync(f_re[1], u_re, P * sizeof(float), hipMemcpyDeviceToDevice, stream);
      hipMemcpyAsync(f_im[1], u_im, P * sizeof(float), hipMemcpyDeviceToDevice, stream);
      dft256_pass<<<G1, 256, 0, stream>>>(fs, 1, 256, 1.0f, INV, LPW1);
      dft256_pass<<<G1, 256, 0, stream>>>(fs, 256, 1, 1.0f, INV, LPW1);
      copy_real_kernel<<<pg, pw, 0, stream>>>(f_re[1], out + (size_t)rec * PLANE_ELEMS);
    }
  }
}